// SpatialGlimpse_15942918603233
// MI455X (gfx1250) — compile-verified
//
#include <hip/hip_runtime.h>

// SpatialGlimpse for MI455X (gfx1250, wave32).
// Separable-stencil reformulation: bilinear + k×k mean-pool == (k+1)-tap
// separable conv with weights [1-f, 1..1, f]/k at stride k, f = frac(center+0.5)
// (identical for all depths). Zero-padded image == reference's OOB masking.
// Bandwidth-bound (~60 MB traffic, ~90 MFLOP) => stage tiles via CDNA5 async
// global->LDS copies (ASYNCcnt path), serve all stencil re-reads from LDS.

#define SG_B     64
#define SG_H     512
#define SG_W     512
#define SG_C     3
#define SG_SIZE  64
#define SG_DEPTH 3
#define SG_ROWF  (SG_W * SG_C)   // 1536 floats per image row

#if defined(__AMDGCN__) && __has_builtin(__builtin_amdgcn_global_load_async_to_lds_b32)
#define SG_HAS_ASYNC 1
#else
#define SG_HAS_ASYNC 0
#endif

#if SG_HAS_ASYNC
typedef __attribute__((address_space(1))) int sg_g32;  // global, 32-bit payload
typedef __attribute__((address_space(3))) int sg_l32;  // LDS (32-bit ptr)

__device__ __forceinline__ void sg_async_copy_b32(const float* gsrc, float* ldst) {
  // Generic LDS pointer = {shared aperture hi32, LDS byte offset lo32}, so the
  // low 32 bits are the DS address; inttoptr casts avoid addrspace-cast Sema.
  __builtin_amdgcn_global_load_async_to_lds_b32(
      (sg_g32*)(unsigned long long)gsrc,
      (sg_l32*)(unsigned int)(unsigned long long)ldst,
      /*imm offset=*/0, /*imm cpol=*/0);
}

__device__ __forceinline__ void sg_wait_async() {
#if __has_builtin(__builtin_amdgcn_s_wait_asynccnt)
  __builtin_amdgcn_s_wait_asynccnt(0);
#else
  asm volatile("s_wait_asynccnt 0" ::: "memory");
#endif
}
#endif  // SG_HAS_ASYNC

__global__ __launch_bounds__(256)
void sg_glimpse_kernel(const float* __restrict__ img,
                       const float* __restrict__ offs,
                       float* __restrict__ out) {
  // Largest phase (d=2): (16*4+1) rows x (16*4+1)*3 floats = 65*195 = 50700 B.
  __shared__ float tile[65 * 195];

  const int tid    = threadIdx.x;
  const int tx     = tid & 15;        // output col within tile
  const int ty     = tid >> 4;        // output row within tile
  const int blk    = blockIdx.x;
  const int b      = blk >> 4;        // batch
  const int t4     = blk & 15;        // 4x4 tile grid over 64x64 output
  const int tile_i = t4 >> 2;
  const int tile_j = t4 & 3;

  // Per-batch center; fractional weight f = frac(c + 0.5) is depth-invariant.
  const float cy  = (offs[2 * b + 0] + 1.0f) * (SG_H * 0.5f);
  const float cx  = (offs[2 * b + 1] + 1.0f) * (SG_W * 0.5f);
  const float syf = cy + 0.5f, sxf = cx + 0.5f;
  const float fly = floorf(syf), flx = floorf(sxf);
  const float fy  = syf - fly,  fx  = sxf - flx;
  const int   iy  = (int)fly,   ix  = (int)flx;

  const long long planeBase = (long long)b * ((long long)SG_H * SG_ROWF);

  float acc[SG_C][SG_DEPTH];

#pragma unroll
  for (int d = 0; d < SG_DEPTH; ++d) {
    const int k      = 1 << d;               // 1, 2, 4
    const int rows   = 16 * k + 1;           // 17, 33, 65 (square tile)
    const int rowf   = rows * SG_C;          // 51, 99, 195 floats per tile row
    const int ystart = iy - 32 * k + tile_i * 16 * k;
    const int xstart = ix - 32 * k + tile_j * 16 * k;
    const int xf0    = xstart * SG_C;        // float index of col 0 within a row
    const int n      = rows * rowf;

    // ---- Stage tile: coalesced async global->LDS; zero-fill OOB via DS. ----
    for (int e = tid; e < n; e += 256) {
      const int r   = e / rowf;
      const int ci  = e - r * rowf;
      const int gy  = ystart + r;
      const int gxf = xf0 + ci;              // bounds multiple of 3 => exact col test
      if (gy >= 0 && gy < SG_H && gxf >= 0 && gxf < SG_ROWF) {
        const float* src = img + planeBase + (long long)gy * SG_ROWF + gxf;
#if SG_HAS_ASYNC
        sg_async_copy_b32(src, &tile[e]);
#else
        tile[e] = *src;
#endif
      } else {
        tile[e] = 0.0f;
      }
    }
#if SG_HAS_ASYNC
    sg_wait_async();                         // this wave's async writes done
#endif
    __syncthreads();                         // drains DScnt + all waves' waits

    // ---- Separable (k+1)-tap weighted sum at stride k, from LDS. ----
    const float invk = 1.0f / (float)k;
    const float wy0 = (1.0f - fy) * invk, wyk = fy * invk;
    const float wx0 = (1.0f - fx) * invk, wxk = fx * invk;

    float a0 = 0.0f, a1 = 0.0f, a2 = 0.0f;
    const int rbase = ty * k;
    const int cbase = tx * k * SG_C;
#pragma unroll
    for (int r = 0; r <= 4; ++r) {
      if (r > k) break;
      const float wr = (r == 0) ? wy0 : ((r == k) ? wyk : invk);
      const int   ro = (rbase + r) * rowf + cbase;
      float s0 = 0.0f, s1 = 0.0f, s2 = 0.0f;
#pragma unroll
      for (int q = 0; q <= 4; ++q) {
        if (q > k) break;
        const float wq = (q == 0) ? wx0 : ((q == k) ? wxk : invk);
        s0 += wq * tile[ro + q * 3 + 0];
        s1 += wq * tile[ro + q * 3 + 1];
        s2 += wq * tile[ro + q * 3 + 2];
      }
      a0 += wr * s0; a1 += wr * s1; a2 += wr * s2;
    }
    acc[0][d] = a0; acc[1][d] = a1; acc[2][d] = a2;
    __syncthreads();                         // tile reused by next depth phase
  }

  // ---- Store: 9 contiguous floats per thread (coalesced per wave). ----
  const int oi = tile_i * 16 + ty;
  const int oj = tile_j * 16 + tx;
  float* op = out + (((long long)b * SG_SIZE + oi) * SG_SIZE + oj) * (SG_C * SG_DEPTH);
#pragma unroll
  for (int c = 0; c < SG_C; ++c)
#pragma unroll
    for (int d = 0; d < SG_DEPTH; ++d)
      op[c * SG_DEPTH + d] = acc[c][d];
}

extern "C" void kernel_launch(void* const* d_in, const int* in_sizes, int n_in,
                              void* d_out, int out_size, void* d_ws, size_t ws_size,
                              hipStream_t stream) {
  (void)in_sizes; (void)n_in; (void)out_size; (void)d_ws; (void)ws_size;
  const float* img  = (const float*)d_in[0];   // [64,512,512,3] f32
  const float* offs = (const float*)d_in[1];   // [64,2] f32
  float*       out  = (float*)d_out;           // [64,64,64,9] f32

  dim3 grid(SG_B * 16);   // 64 batches x (4x4 output tiles)
  dim3 block(256);        // 16x16 outputs, 8 wave32
  sg_glimpse_kernel<<<grid, block, 0, stream>>>(img, offs, out);
}